// PICell_65403761983623
// MI455X (gfx1250) — compile-verified
//
#include <hip/hip_runtime.h>
#include <stdint.h>

// ---------------------------------------------------------------------------
// Play-operator (Prandtl–Ishlinskii) recurrence as a parallel clamp-scan.
//   y_t = clamp(y_{t-1}, a_t, b_t),  a_t = x_t * w,  b_t = a_t + 1
// Clamp composition (first then second) is associative:
//   (A,B) o (a2,b2) = (clamp(A,a2,b2), clamp(B,a2,b2))
// 3 passes: block aggregates -> scan of 4096 aggregates -> apply.
// MI455X: input (64 MiB) stays resident in the 192 MB L2 between pass 1 and
// pass 3; outputs are streamed with NT stores so they don't evict it.
// HBM traffic ~= 64 MiB in + 64 MiB out -> ~5.5 us floor at 23.3 TB/s.
// gfx1250 paths: global_load_async_to_lds_b128 (+ s_wait_asynccnt),
// ds_load_b128 tile reads, NT global stores. WMMA is inapplicable (min/max
// monoid, not linear algebra).
// ---------------------------------------------------------------------------

#define THREADS 256
#define EPT 16                         // elements per thread (sequential run)
#define CHUNK (THREADS * EPT)          // 4096 elements per block
#define ROWS 20                        // 16 data dwords + 4 pad -> 80B rows
#define TILE_SLOTS (THREADS * ROWS)    // 5120 dwords = 20 KB

struct Clamp { float a, b; };

__device__ __forceinline__ Clamp cident() {
  return { -__builtin_inff(), __builtin_inff() };
}
__device__ __forceinline__ Clamp ccomp(Clamp f, Clamp s) { // f first, then s
  return { fminf(fmaxf(f.a, s.a), s.b), fminf(fmaxf(f.b, s.a), s.b) };
}
__device__ __forceinline__ float capply(float y, Clamp c) {
  return fminf(fmaxf(y, c.a), c.b);
}

// Ordered tree reduction over 256 per-thread aggregates (adjacent pairing
// preserves left-to-right order). Result valid in thread 0.
__device__ __forceinline__ Clamp block_reduce(Clamp v, float* sA, float* sB) {
  const int t = threadIdx.x;
  sA[t] = v.a; sB[t] = v.b;
  __syncthreads();
#pragma unroll
  for (int off = 1; off < THREADS; off <<= 1) {
    if ((t & (2 * off - 1)) == 0) {
      Clamp r{sA[t + off], sB[t + off]};
      v = ccomp(v, r);
      sA[t] = v.a; sB[t] = v.b;
    }
    __syncthreads();
  }
  return v;
}

// Inclusive block scan (Hillis-Steele). On return LDS holds all inclusive
// prefixes (post-barrier), so sA[t-1] is thread t's exclusive prefix.
__device__ __forceinline__ Clamp block_scan_inc(Clamp v, float* sA, float* sB) {
  const int t = threadIdx.x;
  sA[t] = v.a; sB[t] = v.b;
  __syncthreads();
#pragma unroll
  for (int off = 1; off < THREADS; off <<= 1) {
    Clamp nv = v;
    if (t >= off) { Clamp p{sA[t - off], sB[t - off]}; nv = ccomp(p, v); }
    __syncthreads();
    sA[t] = nv.a; sB[t] = nv.b; v = nv;
    __syncthreads();
  }
  return v;
}

// Stage a full CHUNK tile via the CDNA5 async-to-LDS engine. Chunk c (16B)
// lands at byte 16*c + 16*(c>>2): element l -> slot l + 4*(l>>4), i.e. thread
// rows of 20 dwords (80B, 16B-aligned) -> float4/ds_load_b128 readable.
__device__ __forceinline__ void load_tile_async_full(const float* __restrict__ x,
                                                     int blockBase, float* tile) {
  const int tid = threadIdx.x;
#pragma unroll
  for (int k = 0; k < EPT / 4; ++k) {
    int c = tid + k * THREADS;                       // chunk id [0,1024)
    uint32_t la = (uint32_t)(uintptr_t)&tile[4 * c + 4 * (c >> 2)];
    uint32_t go = (uint32_t)(blockBase + 4 * c) * 4u;
    asm volatile("global_load_async_to_lds_b128 %0, %1, %2"
                 :: "v"(la), "v"(go), "s"(x) : "memory");
  }
  asm volatile("s_wait_asynccnt 0x0" ::: "memory");
  __syncthreads();
}

// Pull this thread's 16 elements (contiguous, 16B-aligned) into registers.
__device__ __forceinline__ void read_row(const float* tile, float w, float* a) {
  const float4* row = reinterpret_cast<const float4*>(tile + threadIdx.x * ROWS);
#pragma unroll
  for (int q = 0; q < 4; ++q) {
    float4 v = row[q];
    a[4 * q + 0] = v.x * w; a[4 * q + 1] = v.y * w;
    a[4 * q + 2] = v.z * w; a[4 * q + 3] = v.w * w;
  }
}

__device__ __forceinline__ Clamp seq_aggregate(const float* a) {
  Clamp agg{a[0], a[0] + 1.0f};
#pragma unroll
  for (int j = 1; j < EPT; ++j) agg = ccomp(agg, Clamp{a[j], a[j] + 1.0f});
  return agg;
}

// Pass 1: per-block clamp aggregate.
__global__ void pi_pass1(const float* __restrict__ x, const float* __restrict__ kw,
                         float* __restrict__ aggA, float* __restrict__ aggB, int T) {
  __shared__ __align__(16) float tile[TILE_SLOTS];
  __shared__ float sA[THREADS], sB[THREADS];
  const int tid = threadIdx.x;
  const int blockBase = blockIdx.x * CHUNK;
  const float w = kw[0];
  Clamp agg;
  if (blockBase + CHUNK <= T) {                      // full block: fast path
    load_tile_async_full(x, blockBase, tile);
    float a[EPT];
    read_row(tile, w, a);
    agg = seq_aggregate(a);
  } else {                                           // tail block: guarded
    agg = cident();
    const int lbase = blockBase + tid * EPT;
    for (int j = 0; j < EPT; ++j) {
      int e = lbase + j;
      if (e < T) { float xv = x[e] * w; agg = ccomp(agg, Clamp{xv, xv + 1.0f}); }
    }
  }
  Clamp tot = block_reduce(agg, sA, sB);
  if (tid == 0) { aggA[blockIdx.x] = tot.a; aggB[blockIdx.x] = tot.b; }
}

// Pass 2: one block computes exclusive prefixes over NB block aggregates.
__global__ void pi_pass2(const float* __restrict__ aggA, const float* __restrict__ aggB,
                         float* __restrict__ excA, float* __restrict__ excB, int NB) {
  __shared__ float sA[THREADS], sB[THREADS];
  const int tid = threadIdx.x;
  const int S = (NB + THREADS - 1) / THREADS;
  Clamp agg = cident();
  for (int i = 0; i < S; ++i) {
    int idx = tid * S + i;
    if (idx < NB) { Clamp c{aggA[idx], aggB[idx]}; agg = ccomp(agg, c); }
  }
  block_scan_inc(agg, sA, sB);
  Clamp run = (tid > 0) ? Clamp{sA[tid - 1], sB[tid - 1]} : cident();
  for (int i = 0; i < S; ++i) {
    int idx = tid * S + i;
    if (idx < NB) {
      excA[idx] = run.a; excB[idx] = run.b;
      Clamp c{aggA[idx], aggB[idx]};
      run = ccomp(run, c);
    }
  }
}

// Pass 3: within-block scan (tile re-read is L2-resident) and emit y with
// non-temporal stores so the streamed output doesn't evict the input from L2.
__global__ void pi_pass3(const float* __restrict__ inp, const float* __restrict__ kw,
                         const float* __restrict__ excA, const float* __restrict__ excB,
                         float* __restrict__ out, int T) {
  __shared__ __align__(16) float tile[TILE_SLOTS];
  __shared__ float sA[THREADS], sB[THREADS];
  const int tid = threadIdx.x;
  const int b   = blockIdx.x;
  const int blockBase = b * CHUNK;
  const float* x = inp + 1;
  const float w  = kw[0];
  const float y0 = inp[0];

  if (blockBase + CHUNK <= T) {                      // full block: fast path
    load_tile_async_full(x, blockBase, tile);
    float a[EPT];
    read_row(tile, w, a);
    Clamp agg = seq_aggregate(a);
    block_scan_inc(agg, sA, sB);
    Clamp texc = (tid > 0) ? Clamp{sA[tid - 1], sB[tid - 1]} : cident();
    Clamp pre  = ccomp(Clamp{excA[b], excB[b]}, texc);
    float y = capply(y0, pre);
    float* o = out + 1 + blockBase + tid * EPT;
#pragma unroll
    for (int j = 0; j < EPT; ++j) {
      y = fminf(fmaxf(y, a[j]), a[j] + 1.0f);
      __builtin_nontemporal_store(y, o + j);
    }
  } else {                                           // tail block: guarded
    Clamp agg = cident();
    const int lbase = blockBase + tid * EPT;
    for (int j = 0; j < EPT; ++j) {
      int e = lbase + j;
      if (e < T) { float xv = x[e] * w; agg = ccomp(agg, Clamp{xv, xv + 1.0f}); }
    }
    block_scan_inc(agg, sA, sB);
    Clamp texc = (tid > 0) ? Clamp{sA[tid - 1], sB[tid - 1]} : cident();
    Clamp pre  = ccomp(Clamp{excA[b], excB[b]}, texc);
    float y = capply(y0, pre);
    for (int j = 0; j < EPT; ++j) {
      int e = lbase + j;
      if (e < T) {
        float xv = x[e] * w;
        y = fminf(fmaxf(y, xv), xv + 1.0f);
        __builtin_nontemporal_store(y, out + 1 + e);
      }
    }
  }
  if (b == 0 && tid == 0) out[0] = y0;
}

extern "C" void kernel_launch(void* const* d_in, const int* in_sizes, int n_in,
                              void* d_out, int out_size, void* d_ws, size_t ws_size,
                              hipStream_t stream) {
  const float* inp = (const float*)d_in[0];   // [T+1]
  const float* kw  = (const float*)d_in[1];   // scalar weight
  float* out = (float*)d_out;                 // [T+1]
  (void)n_in; (void)out_size; (void)ws_size;

  const int T  = in_sizes[0] - 1;             // 16,777,216
  const int NB = (T + CHUNK - 1) / CHUNK;     // 4096 blocks

  float* aggA = (float*)d_ws;                 // [NB]
  float* aggB = aggA + NB;                    // [NB]
  float* excA = aggB + NB;                    // [NB]
  float* excB = excA + NB;                    // [NB]

  pi_pass1<<<NB, THREADS, 0, stream>>>(inp + 1, kw, aggA, aggB, T);
  pi_pass2<<<1, THREADS, 0, stream>>>(aggA, aggB, excA, excB, NB);
  pi_pass3<<<NB, THREADS, 0, stream>>>(inp, kw, excA, excB, out, T);
}